// NEPDescriptor_16819091931695
// MI455X (gfx1250) — compile-verified
//
#include <hip/hip_runtime.h>
#include <hip/hip_bf16.h>
#include <stdint.h>

#define CUTOFF_F 5.0f
#define NT 4
#define NR 11   /* N_MAX_RADIAL+1  (radial feature channels)  */
#define NA 7    /* N_MAX_ANGULAR+1 (angular feature channels) */
#define BR 11   /* BASIS_RADIAL+1  */
#define BA 9    /* BASIS_ANGULAR+1 */

/* accumulator layout: atom-major row of 84 floats (pad 81->84 = 21 float4)
   offsets within a row:
     RAD: 0..10, AV: 11 + n*3 + k (11..31), S: 32..38, M: 39 + n*6 + m (39..80) */
#define O_RAD 0
#define O_AV  11
#define O_S   32
#define O_M   39
#define NROW  84

/* LDS padded coefficient layout: rows padded to 12 floats (3 x float4) */
#define RROW  12
#define RPAIR (NR * RROW)        /* 132 floats per type-pair (radial)  */
#define APAIR (NA * RROW)        /* 84  floats per type-pair (angular) */
#define LDSR_N (NT * NT * RPAIR) /* 2112 */
#define LDSA_N (NT * NT * APAIR) /* 1344 */

__global__ void nep_zero(float* __restrict__ ws, int n) {
  int i = blockIdx.x * blockDim.x + threadIdx.x;
  if (i < n) ws[i] = 0.0f;
}

__device__ __forceinline__ float dot12(const float* p, const float bb[12]) {
  const float4* q = reinterpret_cast<const float4*>(p);
  float4 a0 = q[0], a1 = q[1], a2 = q[2];
  float f;
  f = a0.x * bb[0];
  f = fmaf(a0.y, bb[1], f);
  f = fmaf(a0.z, bb[2], f);
  f = fmaf(a0.w, bb[3], f);
  f = fmaf(a1.x, bb[4], f);
  f = fmaf(a1.y, bb[5], f);
  f = fmaf(a1.z, bb[6], f);
  f = fmaf(a1.w, bb[7], f);
  f = fmaf(a2.x, bb[8], f);
  f = fmaf(a2.y, bb[9], f);
  f = fmaf(a2.z, bb[10], f);
  f = fmaf(a2.w, bb[11], f);
  return f;
}

struct __align__(4) F3 { float x, y, z; };

__global__ __launch_bounds__(256) void nep_edges(
    const int*   __restrict__ eidx,   /* [2,E] */
    const float* __restrict__ elen,   /* [E]   */
    const float* __restrict__ evec,   /* [E,3] */
    const int*   __restrict__ types,  /* [N]   */
    const float* __restrict__ cr,     /* [4,4,11,11] */
    const float* __restrict__ ca,     /* [4,4,7,9]   */
    float* __restrict__ ws,           /* [N, NROW]   */
    int E, int N)
{
  __shared__ __align__(16) float ldsR[LDSR_N];
  __shared__ __align__(16) float ldsA[LDSA_N];

  /* zero LDS so the pad lanes of each padded row are exactly 0 */
  for (int i = threadIdx.x; i < LDSR_N; i += 256) ldsR[i] = 0.0f;
  for (int i = threadIdx.x; i < LDSA_N; i += 256) ldsA[i] = 0.0f;
  __syncthreads();

  /* CDNA5 async global->LDS staging (per-lane repack into padded rows) */
  for (int i = threadIdx.x; i < NT * NT * NR * BR; i += 256) {
    int pair = i / (NR * BR);
    int rem  = i - pair * (NR * BR);
    int n = rem / BR, b = rem - n * BR;
    unsigned lds_addr = (unsigned)(size_t)&ldsR[pair * RPAIR + n * RROW + b];
    unsigned goff     = (unsigned)(i * 4);
    asm volatile("global_load_async_to_lds_b32 %0, %1, %2"
                 :: "v"(lds_addr), "v"(goff), "s"(cr) : "memory");
  }
  for (int i = threadIdx.x; i < NT * NT * NA * BA; i += 256) {
    int pair = i / (NA * BA);
    int rem  = i - pair * (NA * BA);
    int n = rem / BA, b = rem - n * BA;
    unsigned lds_addr = (unsigned)(size_t)&ldsA[pair * APAIR + n * RROW + b];
    unsigned goff     = (unsigned)(i * 4);
    asm volatile("global_load_async_to_lds_b32 %0, %1, %2"
                 :: "v"(lds_addr), "v"(goff), "s"(ca) : "memory");
  }
  asm volatile("s_wait_asynccnt 0" ::: "memory");
  __syncthreads();

  int tid = blockIdx.x * blockDim.x + threadIdx.x;
  int gstride = gridDim.x * blockDim.x;

  for (int e = tid; e < E; e += gstride) {
    int e2 = e + gstride;
    if (e2 < E) {  /* gfx1250 global_prefetch for the next stream chunk */
      __builtin_prefetch(&elen[e2], 0, 0);
      __builtin_prefetch(&evec[3 * e2], 0, 0);
      __builtin_prefetch(&eidx[e2], 0, 0);
      __builtin_prefetch(&eidx[E + e2], 0, 0);
    }

    int row = eidx[e];
    int col = eidx[E + e];
    float d = elen[e];
    F3 v = reinterpret_cast<const F3*>(evec)[e];
    int pair = types[row] * NT + types[col];

    float s = d * (1.0f / CUTOFF_F);
    float x = 2.0f * s - 1.0f;
    float cut = 0.0f;
    if (s < 1.0f) {
      float denom = 1.0f - s * s;
      cut = expf(1.0f - 1.0f / denom);
    }
    float inv = 1.0f / fmaxf(d, 1e-8f);
    float dx = v.x * inv, dy = v.y * inv, dz = v.z * inv;

    /* Chebyshev basis * envelope, padded to 12 */
    float bb[12];
    bb[0] = cut;
    bb[1] = x * cut;
    float tkm2 = 1.0f, tkm1 = x;
#pragma unroll
    for (int k = 2; k < 11; ++k) {
      float tk = 2.0f * x * tkm1 - tkm2;
      bb[k] = tk * cut;
      tkm2 = tkm1; tkm1 = tk;
    }
    bb[11] = 0.0f;

    /* single base address per edge; component index folds into the
       instruction's immediate offset (0..324 bytes, well inside IOFFSET) */
    float* base = ws + (size_t)row * NROW;

    const float* R = ldsR + pair * RPAIR;
#pragma unroll
    for (int n = 0; n < NR; ++n) {
      float f = dot12(R + n * RROW, bb);
      atomicAdd(base + (O_RAD + n), f);
    }

    const float* A = ldsA + pair * APAIR;
#pragma unroll
    for (int n = 0; n < NA; ++n) {
      float f = dot12(A + n * RROW, bb);   /* pads are zero in LDS */
      atomicAdd(base + (O_AV + n * 3 + 0), f * dx);
      atomicAdd(base + (O_AV + n * 3 + 1), f * dy);
      atomicAdd(base + (O_AV + n * 3 + 2), f * dz);
      atomicAdd(base + (O_S + n), f);
      atomicAdd(base + (O_M + n * 6 + 0), f * dx * dx);
      atomicAdd(base + (O_M + n * 6 + 1), f * dy * dy);
      atomicAdd(base + (O_M + n * 6 + 2), f * dz * dz);
      atomicAdd(base + (O_M + n * 6 + 3), f * dx * dy);
      atomicAdd(base + (O_M + n * 6 + 4), f * dx * dz);
      atomicAdd(base + (O_M + n * 6 + 5), f * dy * dz);
    }
  }
}

__global__ void nep_finish(const float* __restrict__ ws, float* __restrict__ out, int N) {
  int a = blockIdx.x * blockDim.x + threadIdx.x;
  if (a >= N) return;
  const float* r = ws + (size_t)a * NROW;  /* 336B row, 16B aligned -> b128 loads */
  float* o = out + (size_t)a * 25;
#pragma unroll
  for (int n = 0; n < NR; ++n) o[n] = r[O_RAD + n];
#pragma unroll
  for (int n = 0; n < NA; ++n) {
    float ax = r[O_AV + n * 3 + 0];
    float ay = r[O_AV + n * 3 + 1];
    float az = r[O_AV + n * 3 + 2];
    o[11 + n] = ax * ax + ay * ay + az * az;
  }
#pragma unroll
  for (int n = 0; n < NA; ++n) {
    float S  = r[O_S + n];
    float xx = r[O_M + n * 6 + 0];
    float yy = r[O_M + n * 6 + 1];
    float zz = r[O_M + n * 6 + 2];
    float xy = r[O_M + n * 6 + 3];
    float xz = r[O_M + n * 6 + 4];
    float yz = r[O_M + n * 6 + 5];
    /* AT = 1.5*M - 0.5*S*I ; l2 = sum of squares (sym: diag + 2*offdiag) */
    float axx = 1.5f * xx - 0.5f * S;
    float ayy = 1.5f * yy - 0.5f * S;
    float azz = 1.5f * zz - 0.5f * S;
    float axy = 1.5f * xy, axz = 1.5f * xz, ayz = 1.5f * yz;
    o[18 + n] = axx * axx + ayy * ayy + azz * azz
              + 2.0f * (axy * axy + axz * axz + ayz * ayz);
  }
}

extern "C" void kernel_launch(void* const* d_in, const int* in_sizes, int n_in,
                              void* d_out, int out_size, void* d_ws, size_t ws_size,
                              hipStream_t stream) {
  (void)n_in; (void)out_size; (void)ws_size;
  const int*   eidx  = (const int*)d_in[0];
  const float* elen  = (const float*)d_in[1];
  const float* evec  = (const float*)d_in[2];
  /* d_in[3] = num_atoms scalar on device; use host-known sizes instead */
  const int*   types = (const int*)d_in[4];
  const float* cr    = (const float*)d_in[5];
  const float* ca    = (const float*)d_in[6];
  float* out = (float*)d_out;
  float* ws  = (float*)d_ws;

  int E = in_sizes[1];   /* edge_length count */
  int N = in_sizes[4];   /* types count == num_atoms */

  int zn = NROW * N;
  int zb = (zn + 255) / 256;
  nep_zero<<<zb, 256, 0, stream>>>(ws, zn);

  int eb = (E + 255) / 256;
  if (eb > 1024) eb = 1024;   /* grid-stride; amortize LDS staging */
  if (eb < 1) eb = 1;
  nep_edges<<<eb, 256, 0, stream>>>(eidx, elen, evec, types, cr, ca, ws, E, N);

  int fb = (N + 255) / 256;
  nep_finish<<<fb, 256, 0, stream>>>(ws, out, N);
}